// GCN_17506286698969
// MI455X (gfx1250) — compile-verified
//
#include <hip/hip_runtime.h>
#include <hip/hip_bf16.h>
#include <stdint.h>

typedef __attribute__((ext_vector_type(2))) float v2f;
typedef __attribute__((ext_vector_type(8))) float v8f;
typedef __attribute__((ext_vector_type(4))) unsigned int u32x4;
typedef __attribute__((ext_vector_type(8))) int i32x8;
typedef __attribute__((ext_vector_type(4))) int i32x4;

#define HID 256
#define TILE_M 64
#define TILE_N 64
#define TILE_K 32

// ---------------- degree / normalization ----------------

__global__ void init_deg(float* __restrict__ deg, int n) {
    int i = blockIdx.x * blockDim.x + threadIdx.x;
    if (i < n) deg[i] = 1.0f;   // self-loop contribution
}

__global__ void count_deg(const int* __restrict__ col, float* __restrict__ deg, int e) {
    int i = blockIdx.x * blockDim.x + threadIdx.x;
    if (i < e) atomicAdd(&deg[col[i]], 1.0f);
}

__global__ void make_dinv(float* __restrict__ deg, int n) {
    int i = blockIdx.x * blockDim.x + threadIdx.x;
    if (i < n) {
        float d = deg[i];
        deg[i] = (d > 0.0f) ? rsqrtf(d) : 0.0f;
    }
}

// ---------------- dense GEMM: C[nrows,256] = A[nrows,256] @ W[256,256] ----------------
// fp32 WMMA (V_WMMA_F32_16X16X4_F32). 4 waves/block; 64x64 tile.
// A tile: K-pair-interleaved LDS -> single ds_load_b64 per fragment.
// B tile: plain [k][n] LDS, staged by the Tensor Data Mover (TDM).

__global__ __launch_bounds__(128) void gemm_wmma_f32(const float* __restrict__ A,
                                                     const float* __restrict__ W,
                                                     float* __restrict__ C,
                                                     int nrows) {
    __shared__ float As2[TILE_K / 2][TILE_M][2];  // (k,m) at [k>>1][m][k&1]
    __shared__ float Bs[TILE_K][TILE_N];          // plain row-major tile (TDM target)

    const int tid  = threadIdx.x;
    const int lane = tid & 31;
    const int wave = tid >> 5;            // M-subtile 0..3
    const int m0   = blockIdx.x * TILE_M;
    const int n0   = blockIdx.y * TILE_N;
    const int mh   = lane & 15;           // row/col within 16
    const int kh   = lane >> 4;           // K-pair select (0: K,K+1 ; 1: K+2,K+3)

    v8f c0 = {}, c1 = {}, c2 = {}, c3 = {};

    for (int kc = 0; kc < HID; kc += TILE_K) {
        // ---- stage A tile (guarded rows), transposed + K-pair interleaved ----
        for (int idx = tid; idx < TILE_M * TILE_K; idx += 128) {
            int m = idx >> 5;          // / TILE_K
            int k = idx & (TILE_K - 1);
            int gr = m0 + m;
            As2[k >> 1][m][k & 1] = (gr < nrows) ? A[(size_t)gr * HID + kc + k] : 0.0f;
        }

        // ---- stage B tile via Tensor Data Mover (wave 0 issues the DMA) ----
#if __has_builtin(__builtin_amdgcn_tensor_load_to_lds) && __has_builtin(__builtin_amdgcn_s_wait_tensorcnt)
        if (wave == 0) {
            const unsigned ldsB = (unsigned)(uintptr_t)(&Bs[0][0]);   // addr[31:0] == LDS offset
            const unsigned long long ga =
                (unsigned long long)(uintptr_t)(W + (size_t)kc * HID + n0);
            u32x4 g0;
            g0[0] = 1u;                                        // count=1, user desc
            g0[1] = ldsB;                                      // lds_addr
            g0[2] = (unsigned)(ga & 0xFFFFFFFFull);            // global_addr[31:0]
            g0[3] = (unsigned)((ga >> 32) & 0x01FFFFFFull)     // global_addr[56:32]
                  | (2u << 30);                                // type=2 ("image")
            i32x8 g1;
            g1[0] = (int)(2u << 16);        // data_size=4B; no barrier/iterate/pad; mask=0
            g1[1] = (int)(256u << 16);      // tensor_dim0 = 256 (bits 79:48)
            g1[2] = (int)(256u << 16);      // tensor_dim1 = 256 (bits 111:80)
            g1[3] = (int)(64u << 16);       // tile_dim0 = 64   (bits 127:112)
            g1[4] = 32;                     // tile_dim1 = 32   (bits 143:128); tile_dim2=0
            g1[5] = 256;                    // tensor_dim0_stride = 256 (bits 207:160)
            g1[6] = 0;
            g1[7] = 0;                      // tensor_dim1_stride unused (2-D tile)
            i32x4 gz4 = {0, 0, 0, 0};
            i32x8 gz8 = {0, 0, 0, 0, 0, 0, 0, 0};
            __builtin_amdgcn_tensor_load_to_lds(g0, g1, gz4, gz4, gz8, 0);
            __builtin_amdgcn_s_wait_tensorcnt(0);
        }
#else
        for (int idx = tid; idx < TILE_K * TILE_N; idx += 128) {
            int k = idx >> 6;          // / TILE_N
            int n = idx & (TILE_N - 1);
            Bs[k][n] = W[(size_t)(kc + k) * HID + n0 + n];
        }
#endif
        __syncthreads();

#pragma unroll
        for (int kb = 0; kb < TILE_K; kb += 4) {
            const int ka = kb + 2 * kh;
            // A fragment: one aligned 8-byte LDS load (pair interleaved)
            v2f a = *(const v2f*)&As2[(kb >> 1) + kh][wave * 16 + mh][0];
            v2f b0, b1, b2, b3;
            b0.x = Bs[ka + 0][0 * 16 + mh];  b0.y = Bs[ka + 1][0 * 16 + mh];
            b1.x = Bs[ka + 0][1 * 16 + mh];  b1.y = Bs[ka + 1][1 * 16 + mh];
            b2.x = Bs[ka + 0][2 * 16 + mh];  b2.y = Bs[ka + 1][2 * 16 + mh];
            b3.x = Bs[ka + 0][3 * 16 + mh];  b3.y = Bs[ka + 1][3 * 16 + mh];
            c0 = __builtin_amdgcn_wmma_f32_16x16x4_f32(false, a, false, b0, (short)0, c0, false, false);
            c1 = __builtin_amdgcn_wmma_f32_16x16x4_f32(false, a, false, b1, (short)0, c1, false, false);
            c2 = __builtin_amdgcn_wmma_f32_16x16x4_f32(false, a, false, b2, (short)0, c2, false, false);
            c3 = __builtin_amdgcn_wmma_f32_16x16x4_f32(false, a, false, b3, (short)0, c3, false, false);
        }
        __syncthreads();
    }

    // store: VGPR r -> M = r + 8*kh ; lane low nibble -> N
    const int rbase = m0 + wave * 16 + 8 * kh;
#pragma unroll
    for (int r = 0; r < 8; ++r) {
        int row = rbase + r;
        if (row < nrows) {
            float* dst = C + (size_t)row * HID + n0 + mh;
            dst[0 * 16] = c0[r];
            dst[1 * 16] = c1[r];
            dst[2 * 16] = c2[r];
            dst[3 * 16] = c3[r];
        }
    }
}

// ---------------- edge scatter: acc[col] += hw[row] * dinv[row]*dinv[col] ----------------

__global__ void zero_buf(float* __restrict__ p, size_t n) {
    size_t i = (size_t)blockIdx.x * blockDim.x + threadIdx.x;
    if (i < n) p[i] = 0.0f;
}

__global__ void edge_scatter(const int* __restrict__ rows, const int* __restrict__ cols,
                             const float* __restrict__ dinv, const float* __restrict__ hw,
                             float* __restrict__ acc, int nedges) {
    int e = (blockIdx.x * blockDim.x + threadIdx.x) >> 5;
    if (e >= nedges) return;
    const int lane = threadIdx.x & 31;
    const int r = rows[e];
    const int c = cols[e];
    const float nrm = dinv[r] * dinv[c];
    const float* __restrict__ src = hw + (size_t)r * HID;
    float* __restrict__ dst = acc + (size_t)c * HID;
    __builtin_prefetch(src + lane, 0, 1);
#pragma unroll
    for (int j = 0; j < HID / 32; ++j) {
        int f = lane + 32 * j;
        atomicAdd(&dst[f], src[f] * nrm);
    }
}

// ---------------- finalize: out = [relu](acc + dinv^2 * hw + b) ----------------

__global__ void finalize(const float* __restrict__ acc, const float* __restrict__ hw,
                         const float* __restrict__ dinv, const float* __restrict__ bias,
                         float* __restrict__ out, int n, int do_relu) {
    size_t idx = (size_t)blockIdx.x * blockDim.x + threadIdx.x;
    if (idx >= (size_t)n * HID) return;
    int i = (int)(idx >> 8);
    int f = (int)(idx & (HID - 1));
    float d = dinv[i];
    float v = acc[idx] + hw[idx] * d * d + bias[f];
    if (do_relu) v = fmaxf(v, 0.0f);
    out[idx] = v;
}

// ---------------- decoder: out[i] = emb[i,:] . Wout + bout ----------------

__global__ void decode(const float* __restrict__ emb, const float* __restrict__ Wout,
                       const float* __restrict__ bout, float* __restrict__ out, int n) {
    int i = (blockIdx.x * blockDim.x + threadIdx.x) >> 5;
    if (i >= n) return;
    const int lane = threadIdx.x & 31;
    const float* __restrict__ h = emb + (size_t)i * HID;
    float s = 0.0f;
#pragma unroll
    for (int j = 0; j < HID / 32; ++j) s += h[lane + 32 * j] * Wout[lane + 32 * j];
    for (int off = 16; off; off >>= 1) s += __shfl_xor(s, off, 32);
    if (lane == 0) out[i] = s + bout[0];
}

// ---------------- host orchestration ----------------

extern "C" void kernel_launch(void* const* d_in, const int* in_sizes, int n_in,
                              void* d_out, int out_size, void* d_ws, size_t ws_size,
                              hipStream_t stream) {
    const float* x    = (const float*)d_in[0];
    const int*   ei   = (const int*)d_in[1];   // (2,E): [0..E)=row(src), [E..2E)=col(dst)
    const float* W1   = (const float*)d_in[2];
    const float* b1   = (const float*)d_in[3];
    const float* Wh   = (const float*)d_in[4];
    const float* bh   = (const float*)d_in[5];
    const float* W2   = (const float*)d_in[6];
    const float* b2   = (const float*)d_in[7];
    const float* Wout = (const float*)d_in[8];
    const float* bout = (const float*)d_in[9];

    const int N = in_sizes[0] / HID;
    const int E = in_sizes[1] / 2;
    const int* rows = ei;
    const int* cols = ei + E;

    float* ws   = (float*)d_ws;
    float* dinv = ws;                              // N
    float* hw   = dinv + N;                        // N*HID
    float* acc  = hw + (size_t)N * HID;            // N*HID
    float* hbuf = acc + (size_t)N * HID;           // N*HID

    float* out_pred = (float*)d_out;               // N
    float* emb      = out_pred + N;                // N*HID

    const size_t NF = (size_t)N * HID;
    const int TB = 256;
    const int blkN   = (N + TB - 1) / TB;
    const int blkE   = (E + TB - 1) / TB;
    const int blkNF  = (int)((NF + TB - 1) / TB);
    const int blkEdg = (int)(((size_t)E * 32 + TB - 1) / TB);
    const int blkDec = (int)(((size_t)N * 32 + TB - 1) / TB);
    const dim3 gGemm((N + TILE_M - 1) / TILE_M, HID / TILE_N);

    // degree -> dinv
    init_deg<<<blkN, TB, 0, stream>>>(dinv, N);
    count_deg<<<blkE, TB, 0, stream>>>(cols, dinv, E);
    make_dinv<<<blkN, TB, 0, stream>>>(dinv, N);

    // layer 1: x -> hbuf (relu)
    gemm_wmma_f32<<<gGemm, 128, 0, stream>>>(x, W1, hw, N);
    zero_buf<<<blkNF, TB, 0, stream>>>(acc, NF);
    edge_scatter<<<blkEdg, TB, 0, stream>>>(rows, cols, dinv, hw, acc, E);
    finalize<<<blkNF, TB, 0, stream>>>(acc, hw, dinv, b1, hbuf, N, 1);

    // layer 2: hbuf -> hbuf (relu)
    gemm_wmma_f32<<<gGemm, 128, 0, stream>>>(hbuf, Wh, hw, N);
    zero_buf<<<blkNF, TB, 0, stream>>>(acc, NF);
    edge_scatter<<<blkEdg, TB, 0, stream>>>(rows, cols, dinv, hw, acc, E);
    finalize<<<blkNF, TB, 0, stream>>>(acc, hw, dinv, bh, hbuf, N, 1);

    // layer 3: hbuf -> emb (no relu), emb lives in d_out
    gemm_wmma_f32<<<gGemm, 128, 0, stream>>>(hbuf, W2, hw, N);
    zero_buf<<<blkNF, TB, 0, stream>>>(acc, NF);
    edge_scatter<<<blkEdg, TB, 0, stream>>>(rows, cols, dinv, hw, acc, E);
    finalize<<<blkNF, TB, 0, stream>>>(acc, hw, dinv, b2, emb, N, 0);

    // decoder
    decode<<<blkDec, TB, 0, stream>>>(emb, Wout, bout, out_pred, N);
}